// SpikingLayer_44822278701175
// MI455X (gfx1250) — compile-verified
//
#include <hip/hip_runtime.h>

// LIF spiking layer, hidden branch:
//   f = sigmoid(fleak)
//   for t in 0..7: vmem = vmem*f + x[t]; spike = vmem > 1; out[t] = spike; vmem = spike ? 0 : vmem
//
// x: [B*T, H, W, C] = [256, 32, 32, 128] f32, index on dim0 is b*T + t (B=32, T=8).
// Pure streaming / bandwidth-bound: 268 MB single-pass traffic -> ~11.5 us at 23.3 TB/s.
// Strategy: one thread per (neuron, 4-channel vec), full unroll over T so all 8
// global_load_b128 issue up front (t-stride = 512 KiB fits 24-bit IOFFSET), NT
// temporal hints on both loads and stores (zero reuse, don't pollute L2).

typedef __attribute__((ext_vector_type(4))) float v4f;

#define T_STEPS 8
#define VTHR 1.0f
// per-(b,t) slice H*W*C in float4 units: 32*32*128/4 = 32768 (power of two)
#define S4 32768
#define S4_SHIFT 15

__global__ __launch_bounds__(256) void
spiking_lif_kernel(const float* __restrict__ x,
                   const float* __restrict__ fleak,
                   float* __restrict__ out,
                   int n_threads) {
    const int n = blockIdx.x * blockDim.x + threadIdx.x;  // id over (b, spatial/4)
    if (n >= n_threads) return;

    const int b = n >> S4_SHIFT;        // n / S4
    const int r = n & (S4 - 1);         // n % S4
    // base vec4 index of (b, t=0, r):  b*T*S4 + r  =  (b << 18) + r
    const int base = (b << (S4_SHIFT + 3)) + r;

    const v4f* __restrict__ xin = (const v4f*)x;
    v4f* __restrict__ o = (v4f*)out;

    // wave-uniform leak factor f = sigmoid(fleak[0]) — scalar load + few VALU ops
    const float fl = fleak[0];
    const float f = 1.0f / (1.0f + __expf(-fl));

    v4f vmem = {0.0f, 0.0f, 0.0f, 0.0f};

#pragma unroll
    for (int t = 0; t < T_STEPS; ++t) {
        // t*S4*16B = t*512KiB -> folds into the 24-bit signed instruction offset
        v4f xt = __builtin_nontemporal_load(xin + base + t * S4);
        v4f s;
#pragma unroll
        for (int k = 0; k < 4; ++k) {
            float v  = __builtin_fmaf(vmem[k], f, xt[k]);  // leaky integrate
            float sp = (v > VTHR) ? 1.0f : 0.0f;           // fire (Heaviside, strict >)
            vmem[k]  = (v > VTHR) ? 0.0f : v;              // hard reset
            s[k] = sp;
        }
        __builtin_nontemporal_store(s, o + base + t * S4);
    }
}

extern "C" void kernel_launch(void* const* d_in, const int* in_sizes, int n_in,
                              void* d_out, int out_size, void* d_ws, size_t ws_size,
                              hipStream_t stream) {
    const float* x     = (const float*)d_in[0];   // [256,32,32,128] f32
    const float* fleak = (const float*)d_in[1];   // [1] f32
    float* outp        = (float*)d_out;           // [256,32,32,128] f32

    // one thread per float4 per neuron: total_elems / 4 / T
    const int total_elems = in_sizes[0];                    // 33,554,432
    const int n_threads   = total_elems / (4 * T_STEPS);    // 1,048,576
    const int block = 256;                                  // 8 wave32s
    const int grid  = (n_threads + block - 1) / block;      // 4096

    spiking_lif_kernel<<<grid, block, 0, stream>>>(x, fleak, outp, n_threads);
}